// TOGL_89197880803818
// MI455X (gfx1250) — compile-verified
//
#include <hip/hip_runtime.h>

typedef __attribute__((ext_vector_type(2))) float v2f;
typedef __attribute__((ext_vector_type(8))) float v8f;
typedef int v4i __attribute__((vector_size(16)));
typedef unsigned long long u64;

#define N_NODES 50000
#define N_EDGES 200000
#define N_FEAT  512
#define HIDDEN  256
#define N_FILT  4
#define EPAD    262144  // 2^18 >= N_EDGES
#define NPAD    65536   // 2^16 >= N_NODES

// gfx1250 async global->LDS path (ASYNCcnt-tracked, no VGPR staging)
#if defined(__has_builtin)
#if __has_builtin(__builtin_amdgcn_global_load_async_to_lds_b128)
#define USE_ASYNC_LDS 1
#endif
#endif

// builtin signature: (global v4i*, local v4i*, imm offset, imm cpol)
typedef __attribute__((address_space(1))) v4i gv4i;  // global
typedef __attribute__((address_space(3))) v4i lv4i;  // LDS

__device__ __forceinline__ void wait_asynccnt0() {
#if defined(__has_builtin) && __has_builtin(__builtin_amdgcn_s_wait_asynccnt)
  __builtin_amdgcn_s_wait_asynccnt(0);
#else
  asm volatile("s_wait_asynccnt 0x0" ::: "memory");
#endif
}

// float -> order-preserving unsigned key
__device__ __forceinline__ unsigned ordered_key(float f) {
  unsigned u = __float_as_uint(f);
  return u ^ ((unsigned)((int)u >> 31) | 0x80000000u);
}

// ---------------------------------------------------------------------------
// Fused filtration MLP:  vT[f][n] = (relu(X@W1 + b1) @ W2 + b2)[n][f]
// One block = 16 rows of X; 4 waves, each wave computes a 16x64 hidden strip
// with V_WMMA_F32_16X16X4_F32 (fp32-exact). Hidden strip lives only in LDS;
// GEMM2 (HIDDEN x 4) is folded into the epilogue.
// ---------------------------------------------------------------------------
__global__ __launch_bounds__(128, 2)
void togl_filtration_mlp(const float* __restrict__ X, const float* __restrict__ W1,
                         const float* __restrict__ b1, const float* __restrict__ W2,
                         const float* __restrict__ b2, float* __restrict__ vT) {
  __shared__ float As[16 * 516];          // X strip, padded stride (bank-conflict free)
  __shared__ float Hs[16 * 260];          // relu'd hidden strip
  __shared__ float w2s[HIDDEN * N_FILT];  // W2 staged

  const int tid = threadIdx.x;
  const int m0  = blockIdx.x * 16;

  for (int i = tid; i < HIDDEN * N_FILT; i += 128) w2s[i] = W2[i];

#ifdef USE_ASYNC_LDS
  // stream X[m0..m0+15][0..511] straight into LDS: 16B per lane per row,
  // global_load_async_to_lds_b128, drained on ASYNCcnt before the barrier
  #pragma unroll
  for (int i = 0; i < 16; ++i) {
    const float* g = X + (size_t)(m0 + i) * N_FEAT + tid * 4;
    float* l = &As[i * 516 + tid * 4];
    __builtin_amdgcn_global_load_async_to_lds_b128((gv4i*)(void*)g, (lv4i*)(void*)l,
                                                   0, 0);
  }
  wait_asynccnt0();
#else
  // fallback: cooperative float4 load via VGPRs
  #pragma unroll
  for (int i = 0; i < 16; ++i) {
    float4 v = ((const float4*)(X + (size_t)(m0 + i) * N_FEAT))[tid];
    *(float4*)(&As[i * 516 + tid * 4]) = v;
  }
#endif
  __syncthreads();

  const int wave = tid >> 5, lane = tid & 31;
  const int l = lane & 15, hi = lane >> 4;
  const int nbase = wave * 64;

  v8f c0 = {}, c1 = {}, c2 = {}, c3 = {};
  for (int k = 0; k < N_FEAT; k += 4) {
    // A fragment (16x4 f32): lanes0-15 K={k,k+1}, lanes16-31 K={k+2,k+3}
    v2f a = *(const v2f*)(&As[l * 516 + k + 2 * hi]);
    // B fragments (4x16 f32) for 4 N-tiles, read straight from W1 (hot in L2)
    const float* w0  = W1 + (size_t)(k + 2 * hi) * HIDDEN + nbase + l;
    const float* w1r = w0 + HIDDEN;
    v2f b0 = { w0[0],  w1r[0]  };
    v2f bA = { w0[16], w1r[16] };
    v2f bB = { w0[32], w1r[32] };
    v2f bC = { w0[48], w1r[48] };
    c0 = __builtin_amdgcn_wmma_f32_16x16x4_f32(false, a, false, b0, (short)0, c0, false, false);
    c1 = __builtin_amdgcn_wmma_f32_16x16x4_f32(false, a, false, bA, (short)0, c1, false, false);
    c2 = __builtin_amdgcn_wmma_f32_16x16x4_f32(false, a, false, bB, (short)0, c2, false, false);
    c3 = __builtin_amdgcn_wmma_f32_16x16x4_f32(false, a, false, bC, (short)0, c3, false, false);
  }

  // epilogue 1: + b1, relu, park hidden strip in LDS
  float bb0 = b1[nbase + l],      bb1 = b1[nbase + 16 + l],
        bb2 = b1[nbase + 32 + l], bb3 = b1[nbase + 48 + l];
  #pragma unroll
  for (int r = 0; r < 8; ++r) {
    int row = r + 8 * hi;                 // C layout: VGPR r -> M = r + 8*hi, N = l
    float* h = &Hs[row * 260 + nbase + l];
    h[0]  = fmaxf(c0[r] + bb0, 0.f);
    h[16] = fmaxf(c1[r] + bb1, 0.f);
    h[32] = fmaxf(c2[r] + bb2, 0.f);
    h[48] = fmaxf(c3[r] + bb3, 0.f);
  }
  __syncthreads();

  // epilogue 2: tiny GEMM2 (16x256 @ 256x4) + b2, store transposed filtration
  if (tid < 64) {
    int r = tid >> 2, f = tid & 3;
    float acc = b2[f];
    const float* hrow = &Hs[r * 260];
    #pragma unroll 8
    for (int j = 0; j < HIDDEN; ++j) acc = fmaf(hrow[j], w2s[j * N_FILT + f], acc);
    vT[(size_t)f * N_NODES + m0 + r] = acc;
  }
}

// ---------------------------------------------------------------------------
// Edge filtration values + sortable 64-bit keys (stable: key<<32 | index)
// ---------------------------------------------------------------------------
__global__ void edge_keys(const float* __restrict__ vT, const int* __restrict__ src,
                          const int* __restrict__ dst, float* __restrict__ wgt,
                          u64* __restrict__ ekeys) {
  int i = blockIdx.x * blockDim.x + threadIdx.x;
  if (i >= EPAD) return;
  #pragma unroll
  for (int f = 0; f < N_FILT; ++f) {
    u64 kk;
    if (i < N_EDGES) {
      int u = src[i], t = dst[i];
      float wv = fmaxf(vT[(size_t)f * N_NODES + u], vT[(size_t)f * N_NODES + t]);
      wgt[(size_t)f * N_EDGES + i] = wv;
      kk = ((u64)ordered_key(wv) << 32) | (unsigned)i;
    } else {
      kk = ~0ull;  // padding sinks to the end
    }
    ekeys[(size_t)f * EPAD + i] = kk;
  }
}

__global__ void node_keys(const float* __restrict__ vT, u64* __restrict__ nkeys) {
  int i = blockIdx.x * blockDim.x + threadIdx.x;
  if (i >= NPAD) return;
  #pragma unroll
  for (int f = 0; f < N_FILT; ++f) {
    u64 kk = (i < N_NODES)
      ? (((u64)ordered_key(vT[(size_t)f * N_NODES + i]) << 32) | (unsigned)i)
      : ~0ull;
    nkeys[(size_t)f * NPAD + i] = kk;
  }
}

// one compare-exchange pass of a global bitonic sort; blockIdx.y = filtration
__global__ void bitonic_step(u64* __restrict__ data, int len, int j, int k) {
  int i = blockIdx.x * blockDim.x + threadIdx.x;
  if (i >= len) return;
  u64* d = data + (size_t)blockIdx.y * len;
  int ixj = i ^ j;
  if (ixj > i) {
    u64 a = d[i], b = d[ixj];
    bool up = ((i & k) == 0);
    if ((up && a > b) || (!up && a < b)) { d[i] = b; d[ixj] = a; }
  }
}

// rank[node] = position in ascending value order; maxv = max filtration value
__global__ void node_rank(const u64* __restrict__ nkeys, const float* __restrict__ vT,
                          int* __restrict__ rank, float* __restrict__ maxv) {
  int r = blockIdx.x * blockDim.x + threadIdx.x;
  if (r >= N_NODES) return;
  #pragma unroll
  for (int f = 0; f < N_FILT; ++f) {
    int node = (int)(unsigned)(nkeys[(size_t)f * NPAD + r] & 0xFFFFFFFFull);
    rank[(size_t)f * N_NODES + node] = r;
    if (r == N_NODES - 1) maxv[f] = vT[(size_t)f * N_NODES + node];
  }
}

__global__ void uf_init(const int* __restrict__ rank, int* __restrict__ parent,
                        int* __restrict__ csize, int* __restrict__ mrank,
                        int* __restrict__ mnode, int* __restrict__ kill) {
  int n = blockIdx.x * blockDim.x + threadIdx.x;
  if (n >= N_NODES) return;
  #pragma unroll
  for (int f = 0; f < N_FILT; ++f) {
    size_t o = (size_t)f * N_NODES + n;
    parent[o] = n;
    csize[o]  = 1;
    mrank[o]  = rank[o];
    mnode[o]  = n;
    kill[o]   = -1;
  }
}

// serial Kruskal-style union-find per filtration (the lax.scan); 4 walkers
__global__ void uf_run(const u64* __restrict__ ekeys, const int* __restrict__ src,
                       const int* __restrict__ dst, int* __restrict__ parent,
                       int* __restrict__ csize, int* __restrict__ mrank,
                       int* __restrict__ mnode, int* __restrict__ kill) {
  if (threadIdx.x != 0) return;
  const int f = blockIdx.x;
  int* P  = parent + (size_t)f * N_NODES;
  int* SZ = csize  + (size_t)f * N_NODES;
  int* MR = mrank  + (size_t)f * N_NODES;
  int* MN = mnode  + (size_t)f * N_NODES;
  int* KL = kill   + (size_t)f * N_NODES;
  const u64* ord = ekeys + (size_t)f * EPAD;
  for (int i = 0; i < N_EDGES; ++i) {
    if ((i & 63) == 0)  // latency-bound serial scan: pull the key stream ahead
      __builtin_prefetch(&ord[i + 64], 0, 0);
    int e = (int)(unsigned)(ord[i] & 0xFFFFFFFFull);
    int u = src[e], t = dst[e];
    int ru = u; while (P[ru] != ru) ru = P[ru];  // root is fixed point (== 18-step jump)
    int rt = t; while (P[rt] != rt) rt = P[rt];
    if (ru == rt) continue;
    int young = (MR[ru] > MR[rt]) ? ru : rt;     // younger = larger min birth-rank
    KL[MN[young]] = e;
    int big, small;
    if (SZ[ru] >= SZ[rt]) { big = ru; small = rt; } else { big = rt; small = ru; }
    P[small] = big;
    SZ[big]  = SZ[ru] + SZ[rt];
    int nmr, nmn;
    if (MR[ru] <= MR[rt]) { nmr = MR[ru]; nmn = MN[ru]; }
    else                  { nmr = MR[rt]; nmn = MN[rt]; }
    MR[big] = nmr; MN[big] = nmn;
  }
}

__global__ void finalize(const float* __restrict__ vT, const float* __restrict__ wgt,
                         const int* __restrict__ kill, const float* __restrict__ maxv,
                         float* __restrict__ out) {
  int i = blockIdx.x * blockDim.x + threadIdx.x;
  if (i >= N_FILT * N_NODES) return;
  int f = i / N_NODES;
  int k = kill[i];
  float death = (k >= 0) ? wgt[(size_t)f * N_EDGES + k] : maxv[f];
  out[2 * i]     = vT[i];   // birth
  out[2 * i + 1] = death;
}

// ---------------------------------------------------------------------------
extern "C" void kernel_launch(void* const* d_in, const int* in_sizes, int n_in,
                              void* d_out, int out_size, void* d_ws, size_t ws_size,
                              hipStream_t stream) {
  (void)in_sizes; (void)n_in; (void)out_size; (void)ws_size;
  const float* X  = (const float*)d_in[0];
  const float* W1 = (const float*)d_in[1];
  const float* b1 = (const float*)d_in[2];
  const float* W2 = (const float*)d_in[3];
  const float* b2 = (const float*)d_in[4];
  const int*   ei = (const int*)d_in[5];
  const int* src = ei;
  const int* dst = ei + N_EDGES;
  float* out = (float*)d_out;

  // workspace carve-up (~19 MB, all offsets 8B-aligned)
  float* vT   = (float*)d_ws;                     // 4*50000
  float* wgt  = vT + (size_t)N_FILT * N_NODES;    // 4*200000
  float* maxv = wgt + (size_t)N_FILT * N_EDGES;   // 16 (padded)
  int* rank   = (int*)(maxv + 16);
  int* parent = rank   + (size_t)N_FILT * N_NODES;
  int* csize  = parent + (size_t)N_FILT * N_NODES;
  int* mrank  = csize  + (size_t)N_FILT * N_NODES;
  int* mnode  = mrank  + (size_t)N_FILT * N_NODES;
  int* kill   = mnode  + (size_t)N_FILT * N_NODES;
  u64* ekeys  = (u64*)(kill + (size_t)N_FILT * N_NODES);
  u64* nkeys  = ekeys + (size_t)N_FILT * EPAD;

  togl_filtration_mlp<<<N_NODES / 16, 128, 0, stream>>>(X, W1, b1, W2, b2, vT);
  edge_keys<<<EPAD / 256, 256, 0, stream>>>(vT, src, dst, wgt, ekeys);
  node_keys<<<NPAD / 256, 256, 0, stream>>>(vT, nkeys);

  for (unsigned kk = 2; kk <= EPAD; kk <<= 1)
    for (unsigned j = kk >> 1; j > 0; j >>= 1)
      bitonic_step<<<dim3(EPAD / 256, N_FILT), 256, 0, stream>>>(ekeys, EPAD, (int)j, (int)kk);
  for (unsigned kk = 2; kk <= NPAD; kk <<= 1)
    for (unsigned j = kk >> 1; j > 0; j >>= 1)
      bitonic_step<<<dim3(NPAD / 256, N_FILT), 256, 0, stream>>>(nkeys, NPAD, (int)j, (int)kk);

  node_rank<<<(N_NODES + 255) / 256, 256, 0, stream>>>(nkeys, vT, rank, maxv);
  uf_init<<<(N_NODES + 255) / 256, 256, 0, stream>>>(rank, parent, csize, mrank, mnode, kill);
  uf_run<<<N_FILT, 32, 0, stream>>>(ekeys, src, dst, parent, csize, mrank, mnode, kill);
  finalize<<<(N_FILT * N_NODES + 255) / 256, 256, 0, stream>>>(vT, wgt, kill, maxv, out);
}